// Generator_62801011802974
// MI455X (gfx1250) — compile-verified
//
#include <hip/hip_runtime.h>
#include <hip/hip_bf16.h>

// Problem constants (fixed by the reference)
#define BATCH      1024
#define ZDIM       128
#define NLIN       16
#define EXP_FEATS  12288          // 3*64*64 features per expert
#define NTILES_N   (EXP_FEATS/16) // 768 N-tiles of 16 features
#define MAX_MTILES 80             // sum ceil(n_e/16) <= 64 + 15 = 79

typedef __attribute__((ext_vector_type(2))) float f32x2;
typedef __attribute__((ext_vector_type(8))) float f32x8;

// ---------------------------------------------------------------------------
// Kernel 1: bucket samples by expert, emit 16-sample tile descriptors.
// One block of 1024 threads (32 wave32 waves).
// ---------------------------------------------------------------------------
__global__ void __launch_bounds__(1024)
build_tiles_kernel(const int* __restrict__ c,
                   int* __restrict__ sampleIdx,   // [1024]
                   int* __restrict__ tileBase,    // [MAX_MTILES]
                   int* __restrict__ tileValid,   // [MAX_MTILES]
                   int* __restrict__ tileExpert,  // [MAX_MTILES]
                   int* __restrict__ nTiles) {
    __shared__ int cnt[NLIN];
    __shared__ int off[NLIN];
    int t = threadIdx.x;
    if (t < NLIN) cnt[t] = 0;
    __syncthreads();

    int e = (int)c[t];
    int rank = atomicAdd(&cnt[e], 1);
    __syncthreads();

    if (t == 0) {
        int acc = 0;
        for (int i = 0; i < NLIN; ++i) { off[i] = acc; acc += cnt[i]; }
        int nt = 0;
        for (int i = 0; i < NLIN; ++i) {
            for (int b0 = 0; b0 < cnt[i]; b0 += 16) {
                tileBase[nt]   = off[i] + b0;
                tileValid[nt]  = (cnt[i] - b0) < 16 ? (cnt[i] - b0) : 16;
                tileExpert[nt] = i;
                ++nt;
            }
        }
        *nTiles = nt;
    }
    __syncthreads();
    sampleIdx[off[e] + rank] = t;
}

// ---------------------------------------------------------------------------
// Kernel 2: one wave per (16-sample M-tile, 16-feature N-tile).
// D(16x16) = A(16x4) x B(4x16) + C, iterated 32x over K=128, fp32 WMMA.
// ---------------------------------------------------------------------------
__global__ void __launch_bounds__(32)
moe_wmma_kernel(const float* __restrict__ z,     // [1024,128]
                const float* __restrict__ W,     // [196608,128]
                const float* __restrict__ bias,  // [196608]
                const int*  __restrict__ sampleIdx,
                const int*  __restrict__ tileBase,
                const int*  __restrict__ tileValid,
                const int*  __restrict__ tileExpert,
                const int*  __restrict__ nTiles,
                float* __restrict__ out) {       // [1024,12288]
    int ty = blockIdx.y;
    if (ty >= *nTiles) return;   // uniform per block: EXEC stays all-ones below

    __shared__ int samp[16];
    __shared__ int meta[3];
    int lane = threadIdx.x;
    if (lane == 0) {
        meta[0] = tileBase[ty];
        meta[1] = tileValid[ty];
        meta[2] = tileExpert[ty];
    }
    __syncthreads();
    int base = meta[0], valid = meta[1], expert = meta[2];
    if (lane < 16) {
        int m = lane < valid ? lane : (valid - 1);   // pad with a real sample
        samp[lane] = sampleIdx[base + m];
    }
    __syncthreads();

    int half = lane >> 4;        // 0: K={0,1}  1: K={2,3}
    int lo   = lane & 15;        // A: row M / B: col N
    int kb   = half * 2;

    int fLocal = blockIdx.x * 16 + lo;           // feature within expert slice
    int oRow   = expert * EXP_FEATS + fLocal;    // global W row for this lane's N

    const float* zrow = z + (size_t)samp[lo] * ZDIM + kb;   // A: M = lo
    const float* wrow = W + (size_t)oRow * ZDIM + kb;       // B: N = lo

    f32x8 acc = {};
#pragma unroll
    for (int ki = 0; ki < ZDIM / 4; ++ki) {
        f32x2 a  = *(const f32x2*)(zrow + ki * 4);
        f32x2 bm = *(const f32x2*)(wrow + ki * 4);
        // v_wmma_f32_16x16x4_f32: (neg_a, A, neg_b, B, c_mod, C, reuse_a, reuse_b)
        acc = __builtin_amdgcn_wmma_f32_16x16x4_f32(
                  false, a, false, bm, (short)0, acc, false, false);
    }

    float bl = bias[oRow];
#pragma unroll
    for (int r = 0; r < 8; ++r) {
        int m = r + half * 8;                    // C/D layout: lanes 16-31 -> M+8
        if (m < valid) {
            out[(size_t)samp[m] * EXP_FEATS + fLocal] = acc[r] + bl;
        }
    }
}

extern "C" void kernel_launch(void* const* d_in, const int* in_sizes, int n_in,
                              void* d_out, int out_size, void* d_ws, size_t ws_size,
                              hipStream_t stream) {
    const float* z    = (const float*)d_in[0];
    const int*   c    = (const int*)d_in[1];
    const float* W    = (const float*)d_in[2];
    const float* bias = (const float*)d_in[3];
    float* out        = (float*)d_out;

    int* ws         = (int*)d_ws;
    int* sampleIdx  = ws;                 // 1024
    int* tileBase   = ws + 1024;          // MAX_MTILES
    int* tileValid  = tileBase + MAX_MTILES;
    int* tileExpert = tileValid + MAX_MTILES;
    int* nTiles     = tileExpert + MAX_MTILES;

    build_tiles_kernel<<<1, 1024, 0, stream>>>(c, sampleIdx, tileBase, tileValid,
                                               tileExpert, nTiles);

    dim3 grid(NTILES_N, MAX_MTILES);
    moe_wmma_kernel<<<grid, 32, 0, stream>>>(z, W, bias, sampleIdx, tileBase,
                                             tileValid, tileExpert, nTiles, out);
}